// GCN_1529008357501
// MI455X (gfx1250) — compile-verified
//
#include <hip/hip_runtime.h>
#include <hip/hip_bf16.h>

typedef float v2f __attribute__((ext_vector_type(2)));
typedef float v8f __attribute__((ext_vector_type(8)));

#define HDIM 128

// ---------------------------------------------------------------- utilities
__global__ __launch_bounds__(256) void zero_f32(float* __restrict__ p, long long n) {
  long long t = (long long)blockIdx.x * 256 + threadIdx.x;
  if (t < n) p[t] = 0.0f;
}

// deg[dst] += 1 for each edge
__global__ __launch_bounds__(256) void deg_count(const long long* __restrict__ dst,
                                                 float* __restrict__ deg, long long E) {
  long long e = (long long)blockIdx.x * 256 + threadIdx.x;
  if (e < E) atomicAdd(&deg[(int)dst[e]], 1.0f);
}

// in-place: deg -> rsqrt(deg + 1)   (+1 self loop)
__global__ __launch_bounds__(256) void make_dinv(float* __restrict__ d, long long n) {
  long long i = (long long)blockIdx.x * 256 + threadIdx.x;
  if (i < n) d[i] = rsqrtf(d[i] + 1.0f);
}

// ---------------------------------------------------------------- WMMA GEMM
// O[n,128] = X[n,128] @ W[128,128], fp32 via V_WMMA_F32_16X16X4_F32.
// Block = 256 threads = 8 waves; block handles 16 rows, wave w handles cols [16w,16w+16).
__global__ __launch_bounds__(256) void gemm128_wmma(const float* __restrict__ X,
                                                    const float* __restrict__ W,
                                                    float* __restrict__ O, int n) {
  __shared__ float ldsX[16 * 132];   // padded stride 132 -> conflict-free A reads
  const int tid = threadIdx.x;
  const int rowBase = blockIdx.x * 16;

  // stage the 16x128 X slab into LDS (clamped rows for tail blocks)
  for (int i = tid; i < 16 * 128; i += 256) {
    int r = i >> 7, c = i & 127;
    int row = rowBase + r; if (row >= n) row = n - 1;
    ldsX[r * 132 + c] = X[(long long)row * HDIM + c];
  }
  __syncthreads();

  const int wave = tid >> 5;
  const int lane = tid & 31;
  const int colBase = wave * 16;
  const int m    = lane & 15;          // M (A) / N (B,C,D) index
  const int ksel = (lane >> 4) * 2;    // lane-half K offset

  v8f acc = {};
#pragma unroll
  for (int kk = 0; kk < HDIM; kk += 4) {
    v2f a, b;
    // A fragment: 16x4 f32, lanes 0-15 K={kk,kk+1}, lanes 16-31 K={kk+2,kk+3}
    a.x = ldsX[m * 132 + kk + ksel];
    a.y = ldsX[m * 132 + kk + ksel + 1];
    // B fragment: 4x16 f32, row K striped across lanes within a VGPR
    b.x = W[(kk + ksel) * HDIM + colBase + m];
    b.y = W[(kk + ksel + 1) * HDIM + colBase + m];
    acc = __builtin_amdgcn_wmma_f32_16x16x4_f32(false, a, false, b,
                                                (short)0, acc, false, false);
  }

  // C/D layout: VGPR r -> rows r (lanes 0-15) and r+8 (lanes 16-31), col = lane&15
  const int rowHalf = (lane >> 4) << 3;
#pragma unroll
  for (int r = 0; r < 8; ++r) {
    int row = rowBase + r + rowHalf;
    if (row < n) O[(long long)row * HDIM + colBase + m] = acc[r];
  }
}

// ---------------------------------------------------------------- edge scatter
// one wave per edge; lane handles a float4 feature chunk; L2-hot atomics
__global__ __launch_bounds__(256) void scatter_edges(const float* __restrict__ h,
                                                     float* __restrict__ agg,
                                                     const long long* __restrict__ src,
                                                     const long long* __restrict__ dst,
                                                     const float* __restrict__ dinv,
                                                     long long E) {
  long long t = (long long)blockIdx.x * 256 + threadIdx.x;
  long long e = t >> 5;
  int chunk = (int)(t & 31);
  if (e >= E) return;
  int s = (int)src[e], d = (int)dst[e];
  float norm = dinv[s] * dinv[d];
  float4 v = ((const float4*)(h + (long long)s * HDIM))[chunk];
  float* ap = agg + (long long)d * HDIM + chunk * 4;
  atomicAdd(ap + 0, v.x * norm);
  atomicAdd(ap + 1, v.y * norm);
  atomicAdd(ap + 2, v.z * norm);
  atomicAdd(ap + 3, v.w * norm);
}

// out = relu(agg + h*dinv^2 + bias), written in place over agg
__global__ __launch_bounds__(256) void finalize_layer(const float* __restrict__ h,
                                                      float* __restrict__ agg,
                                                      const float* __restrict__ bias,
                                                      const float* __restrict__ dinv,
                                                      long long n) {
  long long t = (long long)blockIdx.x * 256 + threadIdx.x;
  if (t >= n * HDIM) return;
  long long i = t >> 7;
  int f = (int)(t & 127);
  float di = dinv[i];
  float v = agg[t] + h[t] * di * di + bias[f];
  agg[t] = fmaxf(v, 0.0f);
}

// ---------------------------------------------------------------- pooling
__global__ __launch_bounds__(256) void pool_count(const long long* __restrict__ batch,
                                                  float* __restrict__ cnt, long long n) {
  long long i = (long long)blockIdx.x * 256 + threadIdx.x;
  if (i < n) atomicAdd(&cnt[(int)batch[i]], 1.0f);
}

__global__ __launch_bounds__(256) void pool_sum(const float* __restrict__ h,
                                                const long long* __restrict__ batch,
                                                float* __restrict__ pool, long long n) {
  long long t = (long long)blockIdx.x * 256 + threadIdx.x;
  if (t >= n * 32) return;
  long long i = t >> 5;
  int c = (int)(t & 31);
  int g = (int)batch[i];
  float4 v = ((const float4*)(h + i * HDIM))[c];
  float* p = pool + (long long)g * HDIM + c * 4;
  atomicAdd(p + 0, v.x);
  atomicAdd(p + 1, v.y);
  atomicAdd(p + 2, v.z);
  atomicAdd(p + 3, v.w);
}

// one wave per graph: out[g] = dot(pool[g]/max(cnt,1), Wlin) + blin
__global__ __launch_bounds__(256) void readout(const float* __restrict__ pool,
                                               const float* __restrict__ cnt,
                                               const float* __restrict__ Wlin,
                                               const float* __restrict__ blin,
                                               float* __restrict__ out, int G) {
  int wave = threadIdx.x >> 5, lane = threadIdx.x & 31;
  int g = blockIdx.x * 8 + wave;
  if (g >= G) return;
  float inv = 1.0f / fmaxf(cnt[g], 1.0f);
  float p = 0.0f;
  for (int f = lane; f < HDIM; f += 32) p += pool[(long long)g * HDIM + f] * Wlin[f];
#pragma unroll
  for (int off = 16; off > 0; off >>= 1) p += __shfl_down(p, off, 32);
  if (lane == 0) out[g] = p * inv + blin[0];
}

// ---------------------------------------------------------------- launcher
extern "C" void kernel_launch(void* const* d_in, const int* in_sizes, int n_in,
                              void* d_out, int out_size, void* d_ws, size_t ws_size,
                              hipStream_t stream) {
  const float*     x    = (const float*)d_in[0];
  const float*     W1   = (const float*)d_in[1];
  const float*     b1   = (const float*)d_in[2];
  const float*     W2   = (const float*)d_in[3];
  const float*     b2   = (const float*)d_in[4];
  const float*     Wlin = (const float*)d_in[5];
  const float*     blin = (const float*)d_in[6];
  const long long* ei   = (const long long*)d_in[7];
  const long long* batch= (const long long*)d_in[8];

  const long long n = (long long)in_sizes[0] / HDIM;   // 100000
  const long long E = (long long)in_sizes[7] / 2;      // 1600000
  const int       G = out_size;                        // 512
  const long long* src = ei;
  const long long* dst = ei + E;

  // workspace layout (floats)
  float* bufA = (float*)d_ws;               // [n,128] GEMM output
  float* bufB = bufA + n * HDIM;            // [n,128] aggregation / layer output
  float* dinv = bufB + n * HDIM;            // [n]
  float* pool = dinv + n;                   // [G,128]
  float* cnt  = pool + (long long)G * HDIM; // [G]

  const int ZB = 256;
  auto blocks = [](long long work) { return (unsigned)((work + 255) / 256); };

  // degrees -> dinv
  hipLaunchKernelGGL(zero_f32,  dim3(blocks(n)), dim3(ZB), 0, stream, dinv, n);
  hipLaunchKernelGGL(deg_count, dim3(blocks(E)), dim3(ZB), 0, stream, dst, dinv, E);
  hipLaunchKernelGGL(make_dinv, dim3(blocks(n)), dim3(ZB), 0, stream, dinv, n);

  const unsigned gemmBlocks = (unsigned)((n + 15) / 16);

  // ---- layer 1
  hipLaunchKernelGGL(gemm128_wmma, dim3(gemmBlocks), dim3(256), 0, stream, x, W1, bufA, (int)n);
  hipLaunchKernelGGL(zero_f32, dim3(blocks(n * HDIM)), dim3(ZB), 0, stream, bufB, n * HDIM);
  hipLaunchKernelGGL(scatter_edges, dim3(blocks(E * 32)), dim3(ZB), 0, stream,
                     bufA, bufB, src, dst, dinv, E);
  hipLaunchKernelGGL(finalize_layer, dim3(blocks(n * HDIM)), dim3(ZB), 0, stream,
                     bufA, bufB, b1, dinv, n);          // bufB = relu(layer1)

  // ---- layer 2
  hipLaunchKernelGGL(gemm128_wmma, dim3(gemmBlocks), dim3(256), 0, stream, bufB, W2, bufA, (int)n);
  hipLaunchKernelGGL(zero_f32, dim3(blocks(n * HDIM)), dim3(ZB), 0, stream, bufB, n * HDIM);
  hipLaunchKernelGGL(scatter_edges, dim3(blocks(E * 32)), dim3(ZB), 0, stream,
                     bufA, bufB, src, dst, dinv, E);
  hipLaunchKernelGGL(finalize_layer, dim3(blocks(n * HDIM)), dim3(ZB), 0, stream,
                     bufA, bufB, b2, dinv, n);          // bufB = relu(layer2)

  // ---- mean pool + linear readout
  hipLaunchKernelGGL(zero_f32, dim3(blocks((long long)G * HDIM + G)), dim3(ZB), 0, stream,
                     pool, (long long)G * HDIM + G);    // pool and cnt are contiguous
  hipLaunchKernelGGL(pool_count, dim3(blocks(n)), dim3(ZB), 0, stream, batch, cnt, n);
  hipLaunchKernelGGL(pool_sum, dim3(blocks(n * 32)), dim3(ZB), 0, stream, bufB, batch, pool, n);
  hipLaunchKernelGGL(readout, dim3((G + 7) / 8), dim3(256), 0, stream,
                     pool, cnt, Wlin, blin, (float*)d_out, G);
}